// GATConv_26414048871032
// MI455X (gfx1250) — compile-verified
//
#include <hip/hip_runtime.h>

#define N_NODES 100000
#define N_EDGES 1600000
#define HEADS   8
#define OUT_CH  16
#define IN_CH   64
#define HC      128          // HEADS*OUT_CH
#define NEG_SLOPE 0.2f

typedef __attribute__((ext_vector_type(2))) float v2f;
typedef __attribute__((ext_vector_type(8))) float v8f;

// ---------------------------------------------------------------------------
// Kernel 1: h = x @ W^T  via V_WMMA_F32_16X16X4_F32 (exact fp32 path).
// One wave -> 16 nodes x 128 channels.  A: 16x4 f32 frags (2 VGPR), B: 4x16.
// ---------------------------------------------------------------------------
__global__ __launch_bounds__(256) void gat_gemm(const float* __restrict__ x,
                                                const float* __restrict__ W,
                                                float* __restrict__ h) {
    const int lane = threadIdx.x & 31;
    const int wave = blockIdx.x * (blockDim.x >> 5) + (threadIdx.x >> 5);
    if (wave >= N_NODES / 16) return;          // wave-uniform: EXEC all-1s for WMMA
    const int m0   = wave * 16;
    const int half = lane >> 4;                // 0: K pair {0,1} / M rows 0-7 on store
    const int l    = lane & 15;                // M (for A) / N (for B, C/D)

    // A fragments: a[k].x = x[m0+l][4k + 2*half], a[k].y = +1  (ISA 16x4 layout)
    v2f a[16];
    const float* xrow = x + (size_t)(m0 + l) * IN_CH + half * 2;
#pragma unroll
    for (int k = 0; k < 16; ++k)
        a[k] = *(const v2f*)(xrow + k * 4);

#pragma unroll
    for (int t = 0; t < 8; ++t) {              // 8 tiles of 16 output channels
        v8f c = {};
        // B[k][n] = W[t*16+n][k0+k]; lane holds N=l, VGPR pair holds K per half
        const float* wrow = W + (size_t)(t * 16 + l) * IN_CH + half * 2;
#pragma unroll
        for (int k = 0; k < 16; ++k) {
            v2f b = *(const v2f*)(wrow + k * 4);
            c = __builtin_amdgcn_wmma_f32_16x16x4_f32(false, a[k], false, b,
                                                      (short)0, c, false, false);
        }
        // C/D layout: VGPR r -> M = r + 8*half, N = l
#pragma unroll
        for (int r = 0; r < 8; ++r)
            h[(size_t)(m0 + half * 8 + r) * HC + t * 16 + l] = c[r];
    }
}

// ---------------------------------------------------------------------------
// Kernel 2: per (node, head): attention logit halves, seed e_max with the
// self-loop logit, zero denom and the output accumulator.
// ---------------------------------------------------------------------------
__global__ __launch_bounds__(256) void gat_node_prep(const float* __restrict__ h,
                                                     const float* __restrict__ att_src,
                                                     const float* __restrict__ att_dst,
                                                     float* __restrict__ a_src,
                                                     float* __restrict__ a_dst,
                                                     float* __restrict__ e_max,
                                                     float* __restrict__ denom,
                                                     float* __restrict__ num) {
    int tid = blockIdx.x * blockDim.x + threadIdx.x;
    if (tid >= N_NODES * HEADS) return;
    const int n  = tid >> 3;
    const int hd = tid & 7;
    const float* hp = h + (size_t)n * HC + hd * OUT_CH;
    float s = 0.f, d = 0.f;
#pragma unroll
    for (int c = 0; c < OUT_CH; ++c) {
        float v = hp[c];
        s += v * att_src[hd * OUT_CH + c];
        d += v * att_dst[hd * OUT_CH + c];
    }
    a_src[tid] = s;
    a_dst[tid] = d;
    float e = s + d;
    e = e > 0.f ? e : NEG_SLOPE * e;           // self-loop logit seeds the max
    e_max[tid] = e;
    denom[tid] = 0.f;
    float* np = num + (size_t)n * HC + hd * OUT_CH;
#pragma unroll
    for (int c = 0; c < OUT_CH; ++c) np[c] = 0.f;
}

// monotone float atomic-max (int order for >=0, reversed uint order for <0)
__device__ __forceinline__ void atomicMaxF(float* a, float v) {
    if (__float_as_int(v) >= 0)
        atomicMax((int*)a, __float_as_int(v));
    else
        atomicMin((unsigned int*)a, __float_as_uint(v));
}

// ---------------------------------------------------------------------------
// Kernel 3: segment-max over incoming edges.  One thread per (edge, head).
// ---------------------------------------------------------------------------
__global__ __launch_bounds__(256) void gat_edge_max(const int* __restrict__ src,
                                                    const int* __restrict__ dst,
                                                    const float* __restrict__ a_src,
                                                    const float* __restrict__ a_dst,
                                                    float* __restrict__ e_max) {
    long long tid = (long long)blockIdx.x * blockDim.x + threadIdx.x;
    if (tid >= (long long)N_EDGES * HEADS) return;
    const int e  = (int)(tid >> 3);
    const int hd = (int)(tid & 7);
    const int s  = src[e];
    const int d  = dst[e];
    float v = a_src[s * HEADS + hd] + a_dst[d * HEADS + hd];
    v = v > 0.f ? v : NEG_SLOPE * v;
    atomicMaxF(&e_max[d * HEADS + hd], v);
}

// ---------------------------------------------------------------------------
// Kernel 4: accumulate un-normalized numerator + denominator.
// One thread per (edge, quarter-head): 4 channels each, 32 threads/edge.
// ---------------------------------------------------------------------------
__global__ __launch_bounds__(256) void gat_edge_acc(const int* __restrict__ src,
                                                    const int* __restrict__ dst,
                                                    const float* __restrict__ a_src,
                                                    const float* __restrict__ a_dst,
                                                    const float* __restrict__ e_max,
                                                    const float* __restrict__ h,
                                                    float* __restrict__ denom,
                                                    float* __restrict__ num) {
    long long tid = (long long)blockIdx.x * blockDim.x + threadIdx.x;
    if (tid >= (long long)N_EDGES * 32) return;
    const int e  = (int)(tid >> 5);
    const int q  = (int)(tid & 31);
    const int hd = q >> 2;                     // head = (q*4)/16
    const int j0 = q * 4;                      // flat channel base in [0,128)
    const int s  = src[e];
    const int d  = dst[e];
    float v = a_src[s * HEADS + hd] + a_dst[d * HEADS + hd];
    v = v > 0.f ? v : NEG_SLOPE * v;
    const float w = __expf(v - e_max[d * HEADS + hd]);
    if ((q & 3) == 0) atomicAdd(&denom[d * HEADS + hd], w);
    const float* hp = h   + (size_t)s * HC + j0;
    float*       np = num + (size_t)d * HC + j0;
#pragma unroll
    for (int c = 0; c < 4; ++c)
        atomicAdd(&np[c], w * hp[c]);
}

// ---------------------------------------------------------------------------
// Kernel 5: fold in self-loop, normalize, bias, ReLU (in place on d_out).
// ---------------------------------------------------------------------------
__global__ __launch_bounds__(256) void gat_finalize(const float* __restrict__ h,
                                                    const float* __restrict__ a_src,
                                                    const float* __restrict__ a_dst,
                                                    const float* __restrict__ e_max,
                                                    const float* __restrict__ denom,
                                                    const float* __restrict__ bias,
                                                    float* __restrict__ out) {
    int tid = blockIdx.x * blockDim.x + threadIdx.x;
    if (tid >= N_NODES * HEADS) return;
    const int n  = tid >> 3;
    const int hd = tid & 7;
    float es = a_src[tid] + a_dst[tid];
    es = es > 0.f ? es : NEG_SLOPE * es;
    const float w  = __expf(es - e_max[tid]);          // self-loop weight
    const float dn = denom[tid] + w + 1e-16f;
    const float inv = 1.0f / dn;
    const float* hp = h   + (size_t)n * HC + hd * OUT_CH;
    float*       op = out + (size_t)n * HC + hd * OUT_CH;
#pragma unroll
    for (int c = 0; c < OUT_CH; ++c) {
        float v = (op[c] + w * hp[c]) * inv + bias[hd * OUT_CH + c];
        op[c] = v > 0.f ? v : 0.f;
    }
}

// ---------------------------------------------------------------------------
extern "C" void kernel_launch(void* const* d_in, const int* in_sizes, int n_in,
                              void* d_out, int out_size, void* d_ws, size_t ws_size,
                              hipStream_t stream) {
    const float* x       = (const float*)d_in[0];
    const int*   eidx    = (const int*)  d_in[1];
    const float* W       = (const float*)d_in[2];
    const float* att_src = (const float*)d_in[3];
    const float* att_dst = (const float*)d_in[4];
    const float* bias    = (const float*)d_in[5];
    float*       out     = (float*)d_out;

    const int* src = eidx;                // edge_index[0]
    const int* dst = eidx + N_EDGES;      // edge_index[1]

    // workspace layout (floats): h | a_src | a_dst | e_max | denom  (~64 MB)
    float* ws    = (float*)d_ws;
    float* h     = ws;                                     // N*128
    float* a_src = h     + (size_t)N_NODES * HC;           // N*8
    float* a_dst = a_src + (size_t)N_NODES * HEADS;
    float* e_max = a_dst + (size_t)N_NODES * HEADS;
    float* denom = e_max + (size_t)N_NODES * HEADS;

    // 1) projection GEMM (WMMA): 6250 waves, 8 waves/block
    {
        const int waves  = N_NODES / 16;
        const int blocks = (waves + 7) / 8;
        gat_gemm<<<blocks, 256, 0, stream>>>(x, W, h);
    }
    // 2) node prep
    {
        const int th = N_NODES * HEADS;
        gat_node_prep<<<(th + 255) / 256, 256, 0, stream>>>(h, att_src, att_dst,
                                                            a_src, a_dst, e_max,
                                                            denom, out);
    }
    // 3) edge segment-max
    {
        const long long th = (long long)N_EDGES * HEADS;
        gat_edge_max<<<(int)((th + 255) / 256), 256, 0, stream>>>(src, dst, a_src,
                                                                  a_dst, e_max);
    }
    // 4) edge accumulate
    {
        const long long th = (long long)N_EDGES * 32;
        gat_edge_acc<<<(int)((th + 255) / 256), 256, 0, stream>>>(src, dst, a_src,
                                                                  a_dst, e_max, h,
                                                                  denom, out);
    }
    // 5) finalize
    {
        const int th = N_NODES * HEADS;
        gat_finalize<<<(th + 255) / 256, 256, 0, stream>>>(h, a_src, a_dst, e_max,
                                                           denom, bias, out);
    }
}